// MoogVCFRFModelCell_55035710931664
// MI455X (gfx1250) — compile-verified
//
#include <hip/hip_runtime.h>

// ---------------------------------------------------------------------------
// Moog ladder VCF cell, batched (B = 4.19M). Memory-bound: 48 B / element,
// ~201 MB total -> ~8.6 us floor at 23.3 TB/s HBM. The 4x4 state update is
// folded into V_WMMA_F32_16X16X4_F32 (D = X(16x4) @ Mc^T(4x16)), which lets
// the A-operand loads be contiguous B64 loads straight from the [B,4] state
// array; a 512B-per-wave LDS transpose restores lane-=element layout for
// full-wave B128 stores of x_n.
// ---------------------------------------------------------------------------

typedef __attribute__((ext_vector_type(2))) float v2f;
typedef __attribute__((ext_vector_type(8))) float v8f;

// ws layout (floats): [0..15] Mc row-major, [16..19] bvec, [20] gain
__global__ void moog_setup_kernel(const float* __restrict__ gf,
                                  const float* __restrict__ gr,
                                  const float* __restrict__ gp,
                                  float* __restrict__ ws) {
  if (threadIdx.x != 0 || blockIdx.x != 0) return;
  const float kk = 1.0f / 44100.0f;
  const float w  = 2.0f * 3.14159265358979f * gf[0] * 1000.0f;
  float A[4][4] = {
      {-w, 0.f, 0.f, -4.0f * w * gr[0] * 0.5f},
      { w, -w, 0.f, 0.f},
      {0.f,  w, -w, 0.f},
      {0.f, 0.f,  w, -w}};
  float Mm[4][8];  // augmented [I - kA/2 | I]
  float Mp[4][4];  // I + kA/2
  for (int i = 0; i < 4; ++i)
    for (int j = 0; j < 4; ++j) {
      const float t  = 0.5f * kk * A[i][j];
      const float id = (i == j) ? 1.0f : 0.0f;
      Mm[i][j]     = id - t;
      Mm[i][j + 4] = id;
      Mp[i][j]     = id + t;
    }
  // Gauss-Jordan (matrix is diagonally dominant; no pivoting needed)
  for (int c = 0; c < 4; ++c) {
    const float pinv = 1.0f / Mm[c][c];
    for (int j = 0; j < 8; ++j) Mm[c][j] *= pinv;
    for (int r = 0; r < 4; ++r) {
      if (r == c) continue;
      const float f = Mm[r][c];
      for (int j = 0; j < 8; ++j) Mm[r][j] -= f * Mm[c][j];
    }
  }
  // Mc = inv(I-kA/2) @ (I+kA/2)
  for (int i = 0; i < 4; ++i)
    for (int j = 0; j < 4; ++j) {
      float acc = 0.f;
      for (int k = 0; k < 4; ++k) acc += Mm[i][k + 4] * Mp[k][j];
      ws[4 * i + j] = acc;
    }
  // bvec = inv(I-kA/2) @ kB2, kB2 = (k/2)*[w,0,0,0]^T -> (k/2)w * inv[:,0]
  const float kb = 0.5f * kk * w;
  for (int i = 0; i < 4; ++i) ws[16 + i] = Mm[i][4] * kb;
  ws[20] = gp[0];
}

__global__ __launch_bounds__(256) void moog_cell_kernel(
    const float* __restrict__ u_n, const float* __restrict__ x_n1,
    const float* __restrict__ u_n1, const float* __restrict__ ws,
    float* __restrict__ y_out, float* __restrict__ x_out,
    float* __restrict__ u_out, int batch) {
  __shared__ __align__(16) float lds[8][128];  // 512 B per wave
  const int lane = threadIdx.x & 31;
  const int wv   = threadIdx.x >> 5;
  const int n    = lane & 15;   // WMMA row (A) / column (D) index
  const int h    = lane >> 4;   // lane-half selects K-pair {0,1} vs {2,3}

  // Uniform constants (uniform pointer + constant index -> s_load broadcast)
  const float  gain = ws[20];
  const float4 bv   = *(const float4*)(ws + 16);

  // B operand = Mc^T (4x16, cols>=4 zero). ISA B layout for 16x16x4 f32:
  // v0 = rows K=0 (lanes 0-15) / K=2 (lanes 16-31), v1 = K=1 / K=3.
  v2f bmat; bmat.x = 0.f; bmat.y = 0.f;
  if (n < 4) {                       // B[k][n] = Mc[n][k]
    bmat.x = ws[4 * n + 2 * h];
    bmat.y = ws[4 * n + 2 * h + 1];
  }
  const v8f cz = {0.f, 0.f, 0.f, 0.f, 0.f, 0.f, 0.f, 0.f};

  const long tid      = (long)blockIdx.x * blockDim.x + threadIdx.x;
  const long waveId   = tid >> 5;
  const long nthreads = (long)gridDim.x * blockDim.x;
  const long nwaves   = nthreads >> 5;
  const long batch32  = (long)batch & ~31L;

  for (long base = waveId * 32; base < batch32; base += nwaves * 32) {
    // A operand: lane (n,h) holds x_n1[base+n][2h : 2h+2]  (contiguous 256B
    // per half-group). Two groups of 16 elements per wave iteration.
    const v2f a0 = *(const v2f*)(x_n1 + ((base + n) << 2) + 2 * h);
    const v2f a1 = *(const v2f*)(x_n1 + ((base + 16 + n) << 2) + 2 * h);

    // D[m][n] = sum_k x[base+m][k] * Mc[n][k] = (Mc @ x_{base+m})[n]
    v8f d0 = __builtin_amdgcn_wmma_f32_16x16x4_f32(
        false, a0, false, bmat, (short)0, cz, false, false);
    v8f d1 = __builtin_amdgcn_wmma_f32_16x16x4_f32(
        false, a1, false, bmat, (short)0, cz, false, false);

    // Transpose D (lane=state-col, vgpr=batch-row) -> lane=element via LDS.
    if (n < 4) {
#pragma unroll
      for (int v = 0; v < 8; ++v) {
        lds[wv][(v + 8 * h) * 4 + n]      = d0[v];   // elements base+0..15
        lds[wv][(16 + v + 8 * h) * 4 + n] = d1[v];   // elements base+16..31
      }
    }
    asm volatile("s_wait_dscnt 0" ::: "memory");  // same-wave LDS RAW guard

    const long  e  = base + lane;
    const float4 p = *(const float4*)&lds[wv][lane << 2];
    const float uu = u_n[e];
    const float s  = uu + u_n1[e];
    float4 xo;
    xo.x = fmaf(bv.x, s, p.x);
    xo.y = fmaf(bv.y, s, p.y);
    xo.z = fmaf(bv.z, s, p.z);
    xo.w = fmaf(bv.w, s, p.w);
    *(float4*)(x_out + (e << 2)) = xo;   // full-wave B128, 512B/wave
    y_out[e] = gain * xo.w;
    u_out[e] = uu;
    asm volatile("s_wait_dscnt 0" ::: "memory");  // WAR guard before reuse
  }

  // Scalar tail (batch not a multiple of 32; BATCH=4194304 never hits this)
  if (batch32 < (long)batch) {
    float m[16];
#pragma unroll
    for (int i = 0; i < 16; ++i) m[i] = ws[i];
    for (long e = batch32 + tid; e < (long)batch; e += nthreads) {
      const float4 xi = *(const float4*)(x_n1 + (e << 2));
      const float uu  = u_n[e];
      const float s   = uu + u_n1[e];
      float4 xo;
      xo.x = m[0]  * xi.x + m[1]  * xi.y + m[2]  * xi.z + m[3]  * xi.w + bv.x * s;
      xo.y = m[4]  * xi.x + m[5]  * xi.y + m[6]  * xi.z + m[7]  * xi.w + bv.y * s;
      xo.z = m[8]  * xi.x + m[9]  * xi.y + m[10] * xi.z + m[11] * xi.w + bv.z * s;
      xo.w = m[12] * xi.x + m[13] * xi.y + m[14] * xi.z + m[15] * xi.w + bv.w * s;
      *(float4*)(x_out + (e << 2)) = xo;
      y_out[e] = gain * xo.w;
      u_out[e] = uu;
    }
  }
}

extern "C" void kernel_launch(void* const* d_in, const int* in_sizes, int n_in,
                              void* d_out, int out_size, void* d_ws, size_t ws_size,
                              hipStream_t stream) {
  // setup_inputs order: u_n, x_n1, u_n1, batch_size, gf, gr, gain_p
  const float* u  = (const float*)d_in[0];
  const float* x1 = (const float*)d_in[1];
  const float* u1 = (const float*)d_in[2];
  const float* gf = (const float*)d_in[4];
  const float* gr = (const float*)d_in[5];
  const float* gp = (const float*)d_in[6];
  const int batch = in_sizes[0];  // u_n flat count == B

  float* ws = (float*)d_ws;
  // d_out = concat(y_n [B], x_n [B*4], u_n [B]) in return order
  float* y  = (float*)d_out;
  float* x  = y + (size_t)batch;
  float* uo = y + (size_t)5 * batch;

  moog_setup_kernel<<<1, 1, 0, stream>>>(gf, gr, gp, ws);

  const int threads = 256;                       // 8 wave32 per block
  long waves  = ((long)batch + 31) / 32;
  long blocks = (waves + 7) / 8;
  if (blocks > 2048) blocks = 2048;              // 16K waves, ~8 iters each
  if (blocks < 1) blocks = 1;
  moog_cell_kernel<<<(int)blocks, threads, 0, stream>>>(u, x1, u1, ws, y, x,
                                                        uo, batch);
}